// KGATConv_38001870635852
// MI455X (gfx1250) — compile-verified
//
#include <hip/hip_runtime.h>
#include <math.h>

// CDNA5 (gfx1250) wave32 WMMA types
typedef __attribute__((ext_vector_type(2))) float v2f;
typedef __attribute__((ext_vector_type(8))) float v8f;

#define DFEAT 64
#define NREL  10

__device__ __forceinline__ float lrelu(float x) {
    return x > 0.0f ? x : 0.01f * x;
}

// -----------------------------------------------------------------------------
// Kernel 1: proj[n, r*64+o] = sum_i nfeat[n,i] * rw[r,i,o]
// GEMM [N x 64] x [64 x 640] using V_WMMA_F32_16X16X4_F32.
// Block: 256 threads = 8 waves; each wave owns one 16x16 output tile.
// A tile (16x64) staged in LDS once per block; B (160 KB total) is L2/L0-hot.
// -----------------------------------------------------------------------------
__global__ void proj_gemm_kernel(const float* __restrict__ nfeat,
                                 const float* __restrict__ rw,
                                 float* __restrict__ proj) {
    __shared__ float Alds[16 * DFEAT];

    const int m0  = blockIdx.x * 16;
    const int tid = threadIdx.x;

    // Cooperative A-tile load: 16 rows x 64 cols, one float4 per thread.
    {
        const int row = tid >> 4;           // 0..15
        const int c4  = (tid & 15) << 2;    // 0,4,...,60
        const float4 v = *(const float4*)(nfeat + (size_t)(m0 + row) * DFEAT + c4);
        *(float4*)(&Alds[row * DFEAT + c4]) = v;
    }
    __syncthreads();

    const int wave = tid >> 5;              // 0..7
    const int lane = tid & 31;
    const int half = lane >> 4;             // 0: K pair {0,1}; 1: K pair {2,3}
    const int mrow = lane & 15;             // A row / D column within tile
    const int ntile = blockIdx.y * 8 + wave;        // 0..39
    const int c = ntile * 16 + mrow;                // global column 0..639
    const int r = c >> 6;                           // relation
    const int o = c & 63;                           // output channel
    const float* __restrict__ Bp = rw + (size_t)r * (DFEAT * DFEAT) + o; // B[k][c] = rw[r, k, o]

    v8f acc = {0.f, 0.f, 0.f, 0.f, 0.f, 0.f, 0.f, 0.f};

#pragma unroll
    for (int kk = 0; kk < 16; ++kk) {
        const int kb = kk * 4 + 2 * half;   // this lane's K base
        v2f a, b;
        a.x = Alds[mrow * DFEAT + kb];
        a.y = Alds[mrow * DFEAT + kb + 1];
        b.x = Bp[(size_t)kb * DFEAT];
        b.y = Bp[(size_t)(kb + 1) * DFEAT];
        acc = __builtin_amdgcn_wmma_f32_16x16x4_f32(
            false, a, false, b, (short)0, acc, false, false);
    }

    // D layout: VGPR j -> row (j + 8*half), col = mrow
    float* __restrict__ Dp = proj + (size_t)(m0 + 8 * half) * (NREL * DFEAT)
                                  + (size_t)ntile * 16 + mrow;
#pragma unroll
    for (int j = 0; j < 8; ++j)
        Dp[(size_t)j * (NREL * DFEAT)] = acc[j];
}

// -----------------------------------------------------------------------------
// Kernel 2: per-edge attention logit
//   att[e] = sum_d proj[src,etype,d] * tanh(proj[dst,etype,d] + efeat[e,d])
// -----------------------------------------------------------------------------
__global__ void edge_att_kernel(const float* __restrict__ proj,
                                const float* __restrict__ efeat,
                                const int* __restrict__ src,
                                const int* __restrict__ dst,
                                const int* __restrict__ etype,
                                float* __restrict__ att,
                                int n_edges) {
    const int e = blockIdx.x * blockDim.x + threadIdx.x;
    if (e >= n_edges) return;
    const int r = etype[e];
    const float* __restrict__ tr = proj + ((size_t)src[e] * NREL + r) * DFEAT;
    const float* __restrict__ hr = proj + ((size_t)dst[e] * NREL + r) * DFEAT;
    const float* __restrict__ ef = efeat + (size_t)e * DFEAT;
    float acc = 0.0f;
#pragma unroll 4
    for (int d = 0; d < DFEAT; d += 4) {
        const float4 t = *(const float4*)(tr + d);
        const float4 h = *(const float4*)(hr + d);
        const float4 x = *(const float4*)(ef + d);
        acc += t.x * tanhf(h.x + x.x);
        acc += t.y * tanhf(h.y + x.y);
        acc += t.z * tanhf(h.z + x.z);
        acc += t.w * tanhf(h.w + x.w);
    }
    att[e] = acc;
}

// -----------------------------------------------------------------------------
// Kernel 3: CSR row pointers from sorted dst via per-node lower_bound.
// -----------------------------------------------------------------------------
__global__ void rowptr_kernel(const int* __restrict__ dst,
                              int* __restrict__ row_ptr,
                              int n_nodes, int n_edges) {
    const int v = blockIdx.x * blockDim.x + threadIdx.x;
    if (v > n_nodes) return;
    int lo = 0, hi = n_edges;
    while (lo < hi) {
        const int mid = (lo + hi) >> 1;
        if (dst[mid] < v) lo = mid + 1; else hi = mid;
    }
    row_ptr[v] = lo;
}

// -----------------------------------------------------------------------------
// Kernel 4: per-dst-node edge softmax + weighted aggregation of src features.
// One wave32 per node. Lanes stride edges for max / exp-sum reductions
// (__shfl_xor over 32 lanes), then the 32 lanes carry the 64-dim feature row
// (2 floats/lane) while looping the node's edges serially — nfeat row reads
// are coalesced 256B and L2-resident (25.6 MB << 192 MB L2).
// -----------------------------------------------------------------------------
__global__ void node_softmax_agg_kernel(const float* __restrict__ att,
                                        const int* __restrict__ row_ptr,
                                        const int* __restrict__ src,
                                        const float* __restrict__ nfeat,
                                        float* __restrict__ h_nb,
                                        int n_nodes) {
    const int wid  = (blockIdx.x * blockDim.x + threadIdx.x) >> 5;
    const int lane = threadIdx.x & 31;
    if (wid >= n_nodes) return;

    const int s = row_ptr[wid];
    const int e = row_ptr[wid + 1];

    // segment max
    float m = -INFINITY;
    for (int i = s + lane; i < e; i += 32) m = fmaxf(m, att[i]);
#pragma unroll
    for (int o = 16; o > 0; o >>= 1) m = fmaxf(m, __shfl_xor(m, o, 32));

    // segment sum of exp
    float ssum = 0.0f;
    for (int i = s + lane; i < e; i += 32) ssum += __expf(att[i] - m);
#pragma unroll
    for (int o = 16; o > 0; o >>= 1) ssum += __shfl_xor(ssum, o, 32);
    const float inv = (e > s) ? (1.0f / ssum) : 0.0f;

    // weighted aggregation: lanes hold feature dims (lane, lane+32)
    float a0 = 0.0f, a1 = 0.0f;
    for (int i = s; i < e; ++i) {
        const float w = __expf(att[i] - m) * inv;     // uniform across wave
        const float* __restrict__ row = nfeat + (size_t)src[i] * DFEAT;
        a0 += w * row[lane];
        a1 += w * row[lane + 32];
    }
    h_nb[(size_t)wid * DFEAT + lane]      = a0;
    h_nb[(size_t)wid * DFEAT + lane + 32] = a1;
}

// -----------------------------------------------------------------------------
// Kernel 5: out = lrelu((nfeat + h_nb) @ W1^T) + lrelu((nfeat * h_nb) @ W2^T)
// Fused dual GEMM [N x 64] x [64 x 64] with V_WMMA_F32_16X16X4_F32.
// Block: 128 threads = 4 waves; wave w owns output columns [16w, 16w+16).
// x1/x2 tiles built in LDS. B[k][n] = W[n][k] (W^T), rows of W contiguous.
// -----------------------------------------------------------------------------
__global__ void out_gemm_kernel(const float* __restrict__ nfeat,
                                const float* __restrict__ h_nb,
                                const float* __restrict__ w1,
                                const float* __restrict__ w2,
                                float* __restrict__ out) {
    __shared__ float X1[16 * DFEAT];
    __shared__ float X2[16 * DFEAT];

    const int m0  = blockIdx.x * 16;
    const int tid = threadIdx.x;            // 0..127

    for (int i = tid; i < 16 * DFEAT; i += 128) {
        const int row = i >> 6, col = i & 63;
        const float nf = nfeat[(size_t)(m0 + row) * DFEAT + col];
        const float hb = h_nb[(size_t)(m0 + row) * DFEAT + col];
        X1[i] = nf + hb;
        X2[i] = nf * hb;
    }
    __syncthreads();

    const int wave = tid >> 5;              // 0..3
    const int lane = tid & 31;
    const int half = lane >> 4;
    const int mrow = lane & 15;
    const int ocol = wave * 16 + mrow;      // output channel 0..63

    v8f acc1 = {0.f, 0.f, 0.f, 0.f, 0.f, 0.f, 0.f, 0.f};
    v8f acc2 = {0.f, 0.f, 0.f, 0.f, 0.f, 0.f, 0.f, 0.f};

#pragma unroll
    for (int kk = 0; kk < 16; ++kk) {
        const int kb = kk * 4 + 2 * half;
        v2f a1, a2, b1, b2;
        a1.x = X1[mrow * DFEAT + kb];
        a1.y = X1[mrow * DFEAT + kb + 1];
        a2.x = X2[mrow * DFEAT + kb];
        a2.y = X2[mrow * DFEAT + kb + 1];
        b1.x = w1[(size_t)ocol * DFEAT + kb];       // B1[k][ocol] = W1[ocol][k]
        b1.y = w1[(size_t)ocol * DFEAT + kb + 1];
        b2.x = w2[(size_t)ocol * DFEAT + kb];
        b2.y = w2[(size_t)ocol * DFEAT + kb + 1];
        acc1 = __builtin_amdgcn_wmma_f32_16x16x4_f32(
            false, a1, false, b1, (short)0, acc1, false, false);
        acc2 = __builtin_amdgcn_wmma_f32_16x16x4_f32(
            false, a2, false, b2, (short)0, acc2, false, false);
    }

#pragma unroll
    for (int j = 0; j < 8; ++j) {
        const int m = m0 + j + 8 * half;
        out[(size_t)m * DFEAT + ocol] = lrelu(acc1[j]) + lrelu(acc2[j]);
    }
}

// -----------------------------------------------------------------------------
extern "C" void kernel_launch(void* const* d_in, const int* in_sizes, int n_in,
                              void* d_out, int out_size, void* d_ws, size_t ws_size,
                              hipStream_t stream) {
    const float* nfeat = (const float*)d_in[0];
    const float* efeat = (const float*)d_in[1];
    const float* rw    = (const float*)d_in[2];
    const float* w1    = (const float*)d_in[3];
    const float* w2    = (const float*)d_in[4];
    const int*   src   = (const int*)d_in[5];
    const int*   dst   = (const int*)d_in[6];
    const int*   ety   = (const int*)d_in[7];
    float*       out   = (float*)d_out;

    const int n_nodes = in_sizes[0] / DFEAT;   // 100000 (divisible by 16)
    const int n_edges = in_sizes[5];           // 1250000

    // Workspace carving (proj 256MB, att 5MB, row_ptr, h_nb 25.6MB)
    char*  ws   = (char*)d_ws;
    size_t off  = 0;
    float* proj = (float*)(ws + off); off += (size_t)n_nodes * NREL * DFEAT * sizeof(float);
    float* att  = (float*)(ws + off); off += (size_t)n_edges * sizeof(float);
    int* row_ptr = (int*)(ws + off);  off += (size_t)(n_nodes + 1) * sizeof(int);
    off = (off + 255) & ~(size_t)255;
    float* h_nb = (float*)(ws + off);

    // 1) proj = nfeat @ rw  (WMMA f32, N-tiles split over gridDim.y)
    proj_gemm_kernel<<<dim3(n_nodes / 16, 5), 256, 0, stream>>>(nfeat, rw, proj);

    // 2) per-edge attention logits
    edge_att_kernel<<<(n_edges + 255) / 256, 256, 0, stream>>>(
        proj, efeat, src, dst, ety, att, n_edges);

    // 3) CSR row pointers from sorted dst
    rowptr_kernel<<<(n_nodes + 1 + 255) / 256, 256, 0, stream>>>(
        dst, row_ptr, n_nodes, n_edges);

    // 4) edge softmax + weighted aggregation (one wave32 per node)
    node_softmax_agg_kernel<<<((size_t)n_nodes * 32 + 255) / 256, 256, 0, stream>>>(
        att, row_ptr, src, nfeat, h_nb, n_nodes);

    // 5) fused residual dual-GEMM + leaky_relu (WMMA f32)
    out_gemm_kernel<<<n_nodes / 16, 128, 0, stream>>>(nfeat, h_nb, w1, w2, out);
}